// AEKD_Teacher_42013370090079
// MI455X (gfx1250) — compile-verified
//
#include <hip/hip_runtime.h>
#include <hip/hip_bf16.h>
#include <stdint.h>

#define C_CLASSES 1000
#define TPB 256
#define F4_PER_ROW 250   // 1000 floats / 4 per b128 transfer

// ---- CDNA5 async global->LDS (ASYNCcnt) -------------------------------------
#if defined(__HIP_DEVICE_COMPILE__)
#if __has_builtin(__builtin_amdgcn_global_load_async_to_lds_b128)
#define HAVE_ASYNC_BUILTIN 1
#endif
#if __has_builtin(__builtin_amdgcn_s_wait_asynccnt)
#define HAVE_WAITASYNC_BUILTIN 1
#endif
#endif

typedef int v4i_vs __attribute__((vector_size(16)));

__device__ __forceinline__ void async_ld_b128(const float* g, void* lds_ptr) {
#if defined(HAVE_ASYNC_BUILTIN)
  typedef __attribute__((address_space(1))) v4i_vs gv4i_t;  // global (AS1)
  typedef __attribute__((address_space(3))) v4i_vs lv4i_t;  // LDS (AS3)
  __builtin_amdgcn_global_load_async_to_lds_b128(
      (gv4i_t*)(uintptr_t)g,
      (lv4i_t*)(uint32_t)(uintptr_t)lds_ptr,  // low 32 bits of flat addr == LDS offset
      0, 0);
#else
  uint32_t lo = (uint32_t)(uintptr_t)lds_ptr;
  uint64_t ga = (uint64_t)(uintptr_t)g;
  asm volatile("global_load_async_to_lds_b128 %0, %1, off"
               :: "v"(lo), "v"(ga) : "memory");
#endif
}

template <int NCNT>
__device__ __forceinline__ void wait_asynccnt() {
#if defined(HAVE_WAITASYNC_BUILTIN)
  __builtin_amdgcn_s_wait_asynccnt(NCNT);
#else
  asm volatile("s_wait_asynccnt %0" :: "i"(NCNT) : "memory");
#endif
}

// ---- deterministic per-(sample,model) uniform --------------------------------
__device__ __forceinline__ float u01_hash(uint64_t x) {
  x += 0x9E3779B97F4A7C15ull;
  x = (x ^ (x >> 30)) * 0xBF58476D1CE4E5B9ull;
  x = (x ^ (x >> 27)) * 0x94D049BB133111EBull;
  x ^= x >> 31;
  return (float)((uint32_t)(x >> 40)) * (1.0f / 16777216.0f);  // [0,1)
}

__global__ __launch_bounds__(TPB) void vote_kernel(
    const float* __restrict__ in0, const float* __restrict__ in1,
    const float* __restrict__ in2, const float* __restrict__ in3,
    float* __restrict__ out, int N) {
  __shared__ alignas(16) float buf[2][4 * C_CLASSES];  // 32 KB double-buffered staging
  __shared__ float red_v[4][TPB / 32];
  __shared__ int   red_i[4][TPB / 32];
  __shared__ int   s_sel;
  __shared__ int   s_tie;

  const int t = threadIdx.x;
  const float* inp[4] = {in0, in1, in2, in3};

  int s = (int)blockIdx.x;
  const int stride = (int)gridDim.x;
  int cur = 0;

  // Prologue: stage first sample (4 async b128 per wave; ASYNCcnt += 4)
  if (s < N && t < F4_PER_ROW) {
#pragma unroll
    for (int m = 0; m < 4; ++m)
      async_ld_b128(inp[m] + (size_t)s * C_CLASSES + t * 4,
                    &buf[0][m * C_CLASSES + t * 4]);
  }

  while (s < N) {
    const int nxt = s + stride;
    const bool have_next = (nxt < N);

    // Issue next sample's loads into the alternate buffer (overlap with compute)
    if (have_next && t < F4_PER_ROW) {
#pragma unroll
      for (int m = 0; m < 4; ++m)
        async_ld_b128(inp[m] + (size_t)nxt * C_CLASSES + t * 4,
                      &buf[cur ^ 1][m * C_CLASSES + t * 4]);
    }
    // Async loads complete in order: cnt<=4 retires the current buffer's 4 ops.
    if (have_next) wait_asynccnt<4>(); else wait_asynccnt<0>();
    __syncthreads();  // cross-wave visibility of LDS staging

    const float* b = buf[cur];

    // Per-thread argmax per model: one ds_load_b128 per model per thread.
    // Thread t < 250 owns classes [4t, 4t+3]; ascending scan + strict '>'
    // preserves first-occurrence argmax semantics.
    float bv[4]; int bi[4];
#pragma unroll
    for (int m = 0; m < 4; ++m) { bv[m] = -3.402823466e38f; bi[m] = 0x7FFFFFFF; }
    if (t < F4_PER_ROW) {
#pragma unroll
      for (int m = 0; m < 4; ++m) {
        const float4 v = *reinterpret_cast<const float4*>(&b[m * C_CLASSES + 4 * t]);
        const float e[4] = {v.x, v.y, v.z, v.w};
#pragma unroll
        for (int j = 0; j < 4; ++j)
          if (e[j] > bv[m]) { bv[m] = e[j]; bi[m] = 4 * t + j; }
      }
    }
    // wave32 shuffle reduction; first-occurrence tie-break (lower index wins)
#pragma unroll
    for (int off = 16; off > 0; off >>= 1) {
#pragma unroll
      for (int m = 0; m < 4; ++m) {
        float ov = __shfl_down(bv[m], off, 32);
        int   oi = __shfl_down(bi[m], off, 32);
        if (ov > bv[m] || (ov == bv[m] && oi < bi[m])) { bv[m] = ov; bi[m] = oi; }
      }
    }
    if ((t & 31) == 0) {
      const int w = t >> 5;
#pragma unroll
      for (int m = 0; m < 4; ++m) { red_v[m][w] = bv[m]; red_i[m][w] = bi[m]; }
    }
    __syncthreads();

    if (t == 0) {
      int lab[4];
#pragma unroll
      for (int m = 0; m < 4; ++m) {
        float best = red_v[m][0]; int besti = red_i[m][0];
        for (int w = 1; w < TPB / 32; ++w) {
          float v = red_v[m][w]; int i = red_i[m][w];
          if (v > best || (v == best && i < besti)) { best = v; besti = i; }
        }
        lab[m] = besti;
      }
      int cnt[4];
#pragma unroll
      for (int m = 0; m < 4; ++m) {
        int c = 0;
#pragma unroll
        for (int j = 0; j < 4; ++j) c += (lab[j] == lab[m]);
        cnt[m] = c;
      }
      int mx = cnt[0];
#pragma unroll
      for (int m = 1; m < 4; ++m) mx = cnt[m] > mx ? cnt[m] : mx;
      int tie = 0;
#pragma unroll
      for (int m = 0; m < 4; ++m)
#pragma unroll
        for (int j = 0; j < 4; ++j)
          if (cnt[m] == mx && cnt[j] == mx && lab[m] != lab[j]) tie = 1;
      float bestr = -1.0f; int sel = 0;
#pragma unroll
      for (int m = 0; m < 4; ++m)
        if (cnt[m] == mx) {
          float r = u01_hash(0x2A5F3C1D9E8B7A64ull ^ ((uint64_t)s * 4ull + (uint64_t)m));
          if (r > bestr) { bestr = r; sel = m; }
        }
      s_sel = sel; s_tie = tie;
    }
    __syncthreads();

    // Output phase: LDS b128 reads -> one global_store_b128 per thread.
    // Rows are 4000 B apart from 16B-aligned bases, so float4 access is legal.
    if (t < F4_PER_ROW) {
      float4* o4 = reinterpret_cast<float4*>(out + (size_t)s * C_CLASSES);
      float4 r;
      if (s_tie) {
        const float4 a0 = *reinterpret_cast<const float4*>(&b[0 * C_CLASSES + 4 * t]);
        const float4 a1 = *reinterpret_cast<const float4*>(&b[1 * C_CLASSES + 4 * t]);
        const float4 a2 = *reinterpret_cast<const float4*>(&b[2 * C_CLASSES + 4 * t]);
        const float4 a3 = *reinterpret_cast<const float4*>(&b[3 * C_CLASSES + 4 * t]);
        r.x = 0.25f * (a0.x + a1.x + a2.x + a3.x);
        r.y = 0.25f * (a0.y + a1.y + a2.y + a3.y);
        r.z = 0.25f * (a0.z + a1.z + a2.z + a3.z);
        r.w = 0.25f * (a0.w + a1.w + a2.w + a3.w);
      } else {
        r = *reinterpret_cast<const float4*>(&b[s_sel * C_CLASSES + 4 * t]);
      }
      o4[t] = r;
    }
    __syncthreads();  // buf[cur] fully consumed before it becomes a load target
    cur ^= 1;
    s = nxt;
  }
}

extern "C" void kernel_launch(void* const* d_in, const int* in_sizes, int n_in,
                              void* d_out, int out_size, void* d_ws, size_t ws_size,
                              hipStream_t stream) {
  (void)n_in; (void)out_size; (void)d_ws; (void)ws_size;
  const float* a = (const float*)d_in[0];
  const float* b = (const float*)d_in[1];
  const float* c = (const float*)d_in[2];
  const float* d = (const float*)d_in[3];
  float* out = (float*)d_out;
  const int N = in_sizes[0] / C_CLASSES;
  int blocks = (N + 3) / 4;  // 4 samples per block via grid stride (double-buffered)
  if (blocks < 1) blocks = 1;
  vote_kernel<<<blocks, TPB, 0, stream>>>(a, b, c, d, out, N);
}